// FraudDetectionHybrid_65481071403651
// MI455X (gfx1250) — compile-verified
//
#include <hip/hip_runtime.h>

// ---------------------------------------------------------------------------
// FraudDetectionHybrid chain kernel for MI455X (gfx1250)
//   - streaming, memory/TRANS balanced elementwise chain
//   - CDNA5 async data mover: global_load_async_to_lds_b128 + s_wait_asynccnt
//   - double-buffered 16KB LDS stages, lane-private slots (no barriers needed)
// ---------------------------------------------------------------------------

#define TPB 256   // threads per block (8 wave32)
#define PPT 4     // patches per thread per stage
#define PPS (TPB * PPT)  // 1024 patches per stage = 16KB (float4 each)

struct ChainParams {
    float wc[4];        // conv kernel
    float bc;           // conv bias
    float wl[5][4];     // layer weights (2x2 row-major)
    float bb[5][2];     // layer biases
    float sc[5][2];     // displacement_r
    float sh[5][2];     // displacement_phi
    float wf[2];        // final linear
    float bf;           // final bias
};

__device__ __forceinline__ float fast_sigmoid(float z) {
    // 1/(1+e^{-z}) = rcp(1 + exp2(-z*log2(e)))
    return __builtin_amdgcn_rcpf(1.0f + __builtin_amdgcn_exp2f(-1.4426950408889634f * z));
}

__device__ __forceinline__ float fast_tanh(float z) {
#if __has_builtin(__builtin_amdgcn_tanhf)
    return __builtin_amdgcn_tanhf(z);           // gfx1250 v_tanh_f32
#else
    // tanh(z) = 1 - 2/(e^{2z}+1); inf-safe at both tails via IEEE semantics
    float e = __builtin_amdgcn_exp2f(2.8853900817779268f * z);
    return 1.0f - 2.0f * __builtin_amdgcn_rcpf(e + 1.0f);
#endif
}

__device__ __forceinline__ float patch_chain(float4 d, const ChainParams& p) {
    // conv 2x2 dot + bias -> sigmoid ; input vector = [s, 0]
    float z = fmaf(d.w, p.wc[3], fmaf(d.z, p.wc[2], fmaf(d.y, p.wc[1], fmaf(d.x, p.wc[0], p.bc))));
    float x0 = fast_sigmoid(z);
    float x1 = 0.0f;
#pragma unroll
    for (int l = 0; l < 5; ++l) {
        // y_i = tanh(x0*W[i][0] + x1*W[i][1] + b_i) ; then *scale + shift
        float h0 = fast_tanh(fmaf(x1, p.wl[l][1], fmaf(x0, p.wl[l][0], p.bb[l][0])));
        float h1 = fast_tanh(fmaf(x1, p.wl[l][3], fmaf(x0, p.wl[l][2], p.bb[l][1])));
        x0 = fmaf(h0, p.sc[l][0], p.sh[l][0]);
        x1 = fmaf(h1, p.sc[l][1], p.sh[l][1]);
    }
    return fmaf(x1, p.wf[1], fmaf(x0, p.wf[0], p.bf));
}

// LDS byte offset (wave-relative) of a __shared__ object, via addrspace(3) cast.
__device__ __forceinline__ unsigned lds_byte_off(const float4* ptr) {
    return (unsigned)(unsigned long long)(__attribute__((address_space(3))) const float4*)ptr;
}

// CDNA5 async data mover: 16B global -> LDS, tracked by ASYNCcnt (not LOADcnt).
__device__ __forceinline__ void async_copy_b128(const float4* gsrc, unsigned lds_off) {
    asm volatile("global_load_async_to_lds_b128 %0, %1, off"
                 :
                 : "v"(lds_off), "v"(gsrc)
                 : "memory");
}

__global__ __launch_bounds__(TPB) void fraud_chain_kernel(
    const float4* __restrict__ data,
    const float* __restrict__ Wc, const float* __restrict__ bc,
    const float* __restrict__ Wl, const float* __restrict__ bl,
    const float* __restrict__ scale, const float* __restrict__ shift,
    const float* __restrict__ Wf, const float* __restrict__ bf,
    float* __restrict__ out,
    long long nPatches, int stagesPerBlock, long long stagedPatches)
{
    __shared__ float4 smem[2 * PPS];   // 32KB double buffer (of 320KB WGP pool)
    const int t = threadIdx.x;

    // ---- tiny uniform parameter set: constant indices -> scalar loads ----
    ChainParams p;
#pragma unroll
    for (int k = 0; k < 4; ++k) p.wc[k] = Wc[k];
    p.bc = bc[0];
#pragma unroll
    for (int l = 0; l < 5; ++l) {
#pragma unroll
        for (int k = 0; k < 4; ++k) p.wl[l][k] = Wl[4 * l + k];
        p.bb[l][0] = bl[2 * l];     p.bb[l][1] = bl[2 * l + 1];
        p.sc[l][0] = scale[2 * l];  p.sc[l][1] = scale[2 * l + 1];
        p.sh[l][0] = shift[2 * l];  p.sh[l][1] = shift[2 * l + 1];
    }
    p.wf[0] = Wf[0]; p.wf[1] = Wf[1]; p.bf = bf[0];

    const long long baseStage = (long long)blockIdx.x * stagesPerBlock;

    if (stagesPerBlock > 0) {
        // Prologue: issue stage 0 into buffer 0 (4 async B128 per lane).
        {
            const float4* g = data + (baseStage * PPS + t);
#pragma unroll
            for (int j = 0; j < PPT; ++j)
                async_copy_b128(g + j * TPB, lds_byte_off(&smem[j * TPB + t]));
        }

        for (int i = 0; i < stagesPerBlock; ++i) {
            const long long stage = baseStage + i;
            const int cur = i & 1;

            if (i + 1 < stagesPerBlock) {
                // Issue next stage into the other buffer, then retire current:
                // async loads complete in order, so ASYNCcnt<=4 means the
                // older stage's 4 transfers have landed in LDS.
                const float4* g = data + ((stage + 1) * PPS + t);
#pragma unroll
                for (int j = 0; j < PPT; ++j)
                    async_copy_b128(g + j * TPB, lds_byte_off(&smem[(cur ^ 1) * PPS + j * TPB + t]));
                asm volatile("s_wait_asynccnt 0x4" ::: "memory");
            } else {
                asm volatile("s_wait_asynccnt 0x0" ::: "memory");
            }

            // Compute current stage from LDS (lane-private slots -> no barrier).
#pragma unroll
            for (int j = 0; j < PPT; ++j) {
                float4 d = smem[cur * PPS + j * TPB + t];
                out[stage * PPS + j * TPB + t] = patch_chain(d, p);
            }
        }
    }

    // Tail (empty for B = 4194304, kept for generality): direct streaming path.
    for (long long q = stagedPatches + (long long)blockIdx.x * TPB + t;
         q < nPatches;
         q += (long long)gridDim.x * TPB) {
        out[q] = patch_chain(data[q], p);
    }
}

extern "C" void kernel_launch(void* const* d_in, const int* in_sizes, int n_in,
                              void* d_out, int out_size, void* d_ws, size_t ws_size,
                              hipStream_t stream) {
    const float4* data  = (const float4*)d_in[0];
    const float*  Wc    = (const float*)d_in[1];
    const float*  bc    = (const float*)d_in[2];
    const float*  Wl    = (const float*)d_in[3];
    const float*  bl    = (const float*)d_in[4];
    const float*  scale = (const float*)d_in[5];
    const float*  shift = (const float*)d_in[6];
    const float*  Wf    = (const float*)d_in[7];
    const float*  bf    = (const float*)d_in[8];
    float* out = (float*)d_out;

    const long long nPatches = (long long)in_sizes[0] / 4;   // 4,194,304
    const long long stages   = nPatches / PPS;               // 4096 full 16KB stages
    int blocks, spb;
    if (stages > 0) {
        blocks = (int)((stages < 1024) ? stages : 1024);     // 1024 blocks (8 waves each)
        spb    = (int)(stages / blocks);                     // 4 stages per block
    } else {
        blocks = 1;
        spb    = 0;
    }
    const long long stagedPatches = (long long)blocks * spb * PPS;

    fraud_chain_kernel<<<blocks, TPB, 0, stream>>>(
        data, Wc, bc, Wl, bl, scale, shift, Wf, bf, out,
        nPatches, spb, stagedPatches);
}